// CayleyNet_63668595195985
// MI455X (gfx1250) — compile-verified
//
#include <hip/hip_runtime.h>
#include <hip/hip_bf16.h>
#include <math.h>
#include <stdint.h>

#define NN   50000
#define EE   800000
#define HH   64
#define NHH  (NN*HH)
#define GG   10
#define NPGC 5000
#define KSEL 4500      // ceil(0.9*5000)
#define RRC  3
#define KKI  4
#define OUTD 10

typedef __attribute__((ext_vector_type(2))) float v2f;
typedef __attribute__((ext_vector_type(8))) float v8f;

// ---------------------------------------------------------------- CSR build
__global__ void k_count(const int* __restrict__ row, const int* __restrict__ col,
                        int* cnt_row, int* cnt_col) {
  int e = blockIdx.x * blockDim.x + threadIdx.x;
  if (e >= EE) return;
  atomicAdd(&cnt_row[row[e]], 1);
  atomicAdd(&cnt_col[col[e]], 1);
}

// single-block exclusive scan of n counts -> off[0..n], 1024 threads
__global__ void k_scan(const int* __restrict__ cnt, int* __restrict__ off, int n) {
  __shared__ int sh[1024];
  int t = threadIdx.x;
  int C = (n + 1023) / 1024;
  int b = t * C;
  int s = 0;
  for (int i = 0; i < C; ++i) { int idx = b + i; if (idx < n) s += cnt[idx]; }
  sh[t] = s; __syncthreads();
  for (int d = 1; d < 1024; d <<= 1) {
    int v = (t >= d) ? sh[t - d] : 0;
    __syncthreads();
    sh[t] += v;
    __syncthreads();
  }
  int excl = sh[t] - s;
  for (int i = 0; i < C; ++i) {
    int idx = b + i;
    if (idx < n) { off[idx] = excl; excl += cnt[idx]; }
  }
  if (t == 1023) off[n] = sh[1023];
}

__global__ void k_copy_cur(const int* off_row, const int* off_col,
                           int* cur_row, int* cur_col) {
  int n = blockIdx.x * blockDim.x + threadIdx.x;
  if (n >= NN) return;
  cur_row[n] = off_row[n];
  cur_col[n] = off_col[n];
}

__global__ void k_scatter(const int* __restrict__ row, const int* __restrict__ col,
                          int* cur_row, int* cur_col,
                          int* adj_row, int* adj_col) {
  int e = blockIdx.x * blockDim.x + threadIdx.x;
  if (e >= EE) return;
  int r = row[e], c = col[e];
  int p = atomicAdd(&cur_row[r], 1); adj_row[p] = c;
  int q = atomicAdd(&cur_col[c], 1); adj_col[q] = r;
}

// deterministic ordering: sort each adjacency segment (avg degree ~16)
__global__ void k_sort_adj(const int* __restrict__ off, int* adj) {
  int n = blockIdx.x * blockDim.x + threadIdx.x;
  if (n >= NN) return;
  int b = off[n], e = off[n + 1];
  for (int i = b + 1; i < e; ++i) {
    int v = adj[i]; int j = i - 1;
    while (j >= b && adj[j] > v) { adj[j + 1] = adj[j]; --j; }
    adj[j + 1] = v;
  }
}

// ---------------------------------------------------------------- per-node Cayley coefficients
// tmp_left = 1/(h*l + i) = (hl - i)/(hl^2+1);  b_dia = tmp_left*(hl - i);  jacW = tmp_left*h
__global__ void k_coeffs(const int* __restrict__ off_row,
                         const float* __restrict__ hp, const float* __restrict__ ap, int l,
                         float2* bdia, float2* jacW, float2* boffW) {
  int n = blockIdx.x * blockDim.x + threadIdx.x;
  if (n >= NN) return;
  float h = hp[l], alpha = ap[l];
  float deg = (float)(off_row[n + 1] - off_row[n]);
  float hl  = h * (deg - alpha);
  float inv = 1.0f / (hl * hl + 1.0f);
  float tlr = hl * inv, tli = -inv;
  bdia[n]  = make_float2((hl * hl - 1.0f) * inv, -2.0f * hl * inv);
  jacW[n]  = make_float2(tlr * h,  tli * h);
  boffW[n] = make_float2(-tlr * h, -tli * h);
}

__global__ void k_init(const float* __restrict__ x, float2* y, float* out,
                       const float* __restrict__ c0p, int l) {
  int i = blockIdx.x * blockDim.x + threadIdx.x;
  if (i >= NHH) return;
  float v = x[i];
  y[i] = make_float2(v, 0.0f);
  out[i] = c0p[l] * v;
}

// ---------------------------------------------------------------- gather SpMM kernels
// b_j[d] = boffW[d] * sum_{c in nbrRow(d)} y[c]  +  bdia[d] * y[d]
// one wave per node, lane handles 2 complex channels (float4)
__global__ void k_bj(const float4* __restrict__ y, float4* __restrict__ bj,
                     const float2* __restrict__ boffW, const float2* __restrict__ bdia,
                     const int* __restrict__ off, const int* __restrict__ adj) {
  int wid  = (blockIdx.x * blockDim.x + threadIdx.x) >> 5;
  int lane = threadIdx.x & 31;
  if (wid >= NN) return;
  int b = off[wid], e = off[wid + 1];
  float4 acc = make_float4(0.f, 0.f, 0.f, 0.f);
  for (int i = b; i < e; ++i) {
    int c = adj[i];
    float4 v = y[(size_t)c * 32 + lane];
    acc.x += v.x; acc.y += v.y; acc.z += v.z; acc.w += v.w;
  }
  float2 w  = boffW[wid];
  float2 dw = bdia[wid];
  float4 s  = y[(size_t)wid * 32 + lane];
  float4 r;
  r.x = acc.x * w.x - acc.y * w.y + s.x * dw.x - s.y * dw.y;
  r.y = acc.x * w.y + acc.y * w.x + s.x * dw.y + s.y * dw.x;
  r.z = acc.z * w.x - acc.w * w.y + s.z * dw.x - s.w * dw.y;
  r.w = acc.z * w.y + acc.w * w.x + s.z * dw.y + s.w * dw.x;
  bj[(size_t)wid * 32 + lane] = r;
}

// yout[d] = sum_{s in nbrCol(d)} jacW[s]*yin[s] + bj[d]
__global__ void k_jacobi(const float4* __restrict__ yin, float4* __restrict__ yout,
                         const float4* __restrict__ bj, const float2* __restrict__ jacW,
                         const int* __restrict__ off, const int* __restrict__ adj) {
  int wid  = (blockIdx.x * blockDim.x + threadIdx.x) >> 5;
  int lane = threadIdx.x & 31;
  if (wid >= NN) return;
  int b = off[wid], e = off[wid + 1];
  float4 acc = bj[(size_t)wid * 32 + lane];
  for (int i = b; i < e; ++i) {
    int s = adj[i];
    if (i + 1 < e) {
      int sn = adj[i + 1];
      __builtin_prefetch((const void*)(yin + (size_t)sn * 32 + lane), 0, 3);
    }
    float2 w = jacW[s];
    float4 v = yin[(size_t)s * 32 + lane];
    acc.x += v.x * w.x - v.y * w.y;
    acc.y += v.x * w.y + v.y * w.x;
    acc.z += v.z * w.x - v.w * w.y;
    acc.w += v.z * w.y + v.w * w.x;
  }
  yout[(size_t)wid * 32 + lane] = acc;
}

__global__ void k_upd(const float2* __restrict__ y, float* out,
                      const float* __restrict__ cj, int l, int j) {
  int i = blockIdx.x * blockDim.x + threadIdx.x;
  if (i >= NHH) return;
  float cr = cj[(l * RRC + j) * 2 + 0];
  float ci = cj[(l * RRC + j) * 2 + 1];
  float2 v = y[i];
  out[i] += 2.0f * (cr * v.x - ci * v.y);
}

__global__ void k_relu(const float* __restrict__ in, float* xo) {
  int i = blockIdx.x * blockDim.x + threadIdx.x;
  if (i >= NHH) return;
  xo[i] = fmaxf(in[i], 0.0f);
}

// ---------------------------------------------------------------- top-k pooling
__global__ void k_wnorm(const float* __restrict__ w, float* invn) {
  __shared__ float sh[64];
  int t = threadIdx.x;
  float v = w[t];
  sh[t] = v * v; __syncthreads();
  for (int d = 32; d > 0; d >>= 1) { if (t < d) sh[t] += sh[t + d]; __syncthreads(); }
  if (t == 0) invn[0] = 1.0f / sqrtf(sh[0]);
}

// one wave per node: score + weighted features
__global__ void k_score(const float* __restrict__ x, const float* __restrict__ w,
                        const float* __restrict__ invn, float* s, float* xp) {
  int wid  = (blockIdx.x * blockDim.x + threadIdx.x) >> 5;
  int lane = threadIdx.x & 31;
  if (wid >= NN) return;
  size_t base = (size_t)wid * HH;
  float x0 = x[base + lane], x1 = x[base + 32 + lane];
  float dot = x0 * w[lane] + x1 * w[32 + lane];
  for (int o = 16; o > 0; o >>= 1) dot += __shfl_xor(dot, o, 32);
  float sc = tanhf(dot * invn[0]);
  if (lane == 0) s[wid] = sc;
  xp[base + lane]      = x0 * sc;
  xp[base + 32 + lane] = x1 * sc;
}

// exact k-th largest per graph via in-LDS bitonic sort (5000 padded to 8192).
// The 20KB contiguous score block is staged global->LDS with the CDNA5 async
// LDS DMA path (GLOBAL_LOAD_ASYNC_TO_LDS_B32, tracked by ASYNCcnt).
__global__ void k_thresh(const float* __restrict__ s, float* thr) {
  __shared__ float a[8192];
  int g = blockIdx.x, t = threadIdx.x;
  // sentinel tail via regular DS stores (disjoint from async-written region)
  for (int i = NPGC + t; i < 8192; i += 512) a[i] = -1e30f;
  // async DMA of the real scores into LDS (per-lane LDS byte address = low
  // 32 bits of the generic shared pointer, per flat-aperture truncation rule)
  const float* src = s + g * NPGC;
  for (int i = t; i < NPGC; i += 512) {
    unsigned lds_off = (unsigned)(uintptr_t)(&a[i]);
    unsigned long long gaddr = (unsigned long long)(uintptr_t)(src + i);
    asm volatile("global_load_async_to_lds_b32 %0, %1, off"
                 :: "v"(lds_off), "v"(gaddr) : "memory");
  }
  asm volatile("s_wait_asynccnt 0x0" ::: "memory");
  __syncthreads();
  for (int k = 2; k <= 8192; k <<= 1) {
    for (int j = k >> 1; j > 0; j >>= 1) {
      for (int i = t; i < 8192; i += 512) {
        int ixj = i ^ j;
        if (ixj > i) {
          bool up = ((i & k) == 0);
          float x0 = a[i], x1 = a[ixj];
          if ((x0 > x1) == up) { a[i] = x1; a[ixj] = x0; }
        }
      }
      __syncthreads();
    }
  }
  if (t == 0) thr[g] = a[8192 - KSEL];   // k-th largest (ascending sort)
}

__global__ void k_pool(const float* __restrict__ xp, const float* __restrict__ s,
                       const float* __restrict__ thr, float* pooled) {
  int g = blockIdx.x, t = threadIdx.x;
  int ch = t & 63, part = t >> 6;             // 256 threads = 4 node-strides x 64 ch
  float acc = 0.0f, tg = thr[g];
  for (int n = part; n < NPGC; n += 4) {
    float sc = s[g * NPGC + n];
    if (sc >= tg) acc += xp[((size_t)g * NPGC + n) * HH + ch];
  }
  __shared__ float sh[256];
  sh[t] = acc; __syncthreads();
  if (part == 0) {
    float v = sh[ch] + sh[ch + 64] + sh[ch + 128] + sh[ch + 192];
    pooled[g * HH + ch] = v * (1.0f / (float)KSEL);
  }
}

// pad lin_w [64,10] into zero-filled [64,16] so WMMA B loads are branchless
__global__ void k_padw(const float* __restrict__ w, float* wpad) {
  int i = blockIdx.x * blockDim.x + threadIdx.x;
  if (i >= HH * 16) return;
  int k = i >> 4, n = i & 15;
  wpad[i] = (n < OUTD) ? w[k * OUTD + n] : 0.0f;
}

// ---------------------------------------------------------------- final linear: pooled[10,64] @ lin_w[64,10] + b
// single wave, exact fp32 via v_wmma_f32_16x16x4_f32, K accumulated in 16 steps
__global__ void k_lin(const float* __restrict__ pooled, const float* __restrict__ wpad,
                      const float* __restrict__ bias, float* __restrict__ out) {
  int lane = threadIdx.x;
  int mn   = lane & 15;              // A: M index; B: N index
  int kofs = (lane < 16) ? 0 : 2;    // 32-bit A 16x4: lanes16-31 hold K=2,3
  v8f c = {};
#pragma unroll
  for (int k0 = 0; k0 < HH; k0 += 4) {
    v2f a, b;
    a.x = pooled[mn * HH + k0 + kofs];        // rows 10..15 pre-zeroed
    a.y = pooled[mn * HH + k0 + kofs + 1];
    b.x = wpad[(k0 + kofs) * 16 + mn];        // cols 10..15 pre-zeroed
    b.y = wpad[(k0 + kofs + 1) * 16 + mn];
    c = __builtin_amdgcn_wmma_f32_16x16x4_f32(false, a, false, b,
                                              (short)0, c, false, false);
  }
  int mbase = (lane < 16) ? 0 : 8;
#pragma unroll
  for (int r = 0; r < 8; ++r) {
    int M = mbase + r;
    if (M < OUTD && mn < OUTD) out[M * OUTD + mn] = c[r] + bias[mn];
  }
}

// ---------------------------------------------------------------- launch
static inline size_t al256(size_t x) { return (x + 255) & ~(size_t)255; }

extern "C" void kernel_launch(void* const* d_in, const int* in_sizes, int n_in,
                              void* d_out, int out_size, void* d_ws, size_t ws_size,
                              hipStream_t stream) {
  const float* x0     = (const float*)d_in[0];
  const int*   ei     = (const int*)  d_in[1];   // [2,E]
  const float* hp     = (const float*)d_in[3];
  const float* ap     = (const float*)d_in[4];
  const float* c0p    = (const float*)d_in[5];
  const float* cjp    = (const float*)d_in[6];
  const float* topkw  = (const float*)d_in[7];
  const float* linw   = (const float*)d_in[8];
  const float* linb   = (const float*)d_in[9];
  float* out = (float*)d_out;

  const int* row = ei;
  const int* col = ei + EE;

  // workspace carving
  char* base = (char*)d_ws;
  size_t o = 0;
  auto carve = [&](size_t bytes) { size_t r = o; o = al256(o + bytes); return r; };
  int*    cur_row = (int*)   (base + carve((size_t)NN * 4));
  int*    cur_col = (int*)   (base + carve((size_t)NN * 4));
  int*    off_row = (int*)   (base + carve((size_t)(NN + 1) * 4));
  int*    off_col = (int*)   (base + carve((size_t)(NN + 1) * 4));
  int*    adj_row = (int*)   (base + carve((size_t)EE * 4));
  int*    adj_col = (int*)   (base + carve((size_t)EE * 4));
  float2* bdia    = (float2*)(base + carve((size_t)NN * 8));
  float2* jacW    = (float2*)(base + carve((size_t)NN * 8));
  float2* boffW   = (float2*)(base + carve((size_t)NN * 8));
  float2* Y       = (float2*)(base + carve((size_t)NHH * 8));
  float2* BJ      = (float2*)(base + carve((size_t)NHH * 8));
  float2* YK2     = (float2*)(base + carve((size_t)NHH * 8));
  float*  xbuf    = (float*) (base + carve((size_t)NHH * 4));
  float*  outbuf  = (float*) (base + carve((size_t)NHH * 4));   // doubles as xp
  float*  sbuf    = (float*) (base + carve((size_t)NN * 4));
  float*  thr     = (float*) (base + carve((size_t)GG * 4));
  float*  pooled  = (float*) (base + carve((size_t)16 * HH * 4));
  float*  wpad    = (float*) (base + carve((size_t)HH * 16 * 4));
  float*  invn    = (float*) (base + carve(4));

  const int TB = 256;
  const int gE  = (EE + TB - 1) / TB;
  const int gN  = (NN + TB - 1) / TB;
  const int gNH = (NHH + TB - 1) / TB;
  const int gW  = (NN * 32 + TB - 1) / TB;   // wave-per-node kernels

  // --- CSR build (deterministic: segments sorted afterwards) ---
  hipMemsetAsync(cur_row, 0, (size_t)NN * 4, stream);
  hipMemsetAsync(cur_col, 0, (size_t)NN * 4, stream);
  k_count<<<gE, TB, 0, stream>>>(row, col, cur_row, cur_col);
  k_scan<<<1, 1024, 0, stream>>>(cur_row, off_row, NN);
  k_scan<<<1, 1024, 0, stream>>>(cur_col, off_col, NN);
  k_copy_cur<<<gN, TB, 0, stream>>>(off_row, off_col, cur_row, cur_col);
  k_scatter<<<gE, TB, 0, stream>>>(row, col, cur_row, cur_col, adj_row, adj_col);
  k_sort_adj<<<gN, TB, 0, stream>>>(off_row, adj_row);
  k_sort_adj<<<gN, TB, 0, stream>>>(off_col, adj_col);

  // --- 2 CayleyConv layers ---
  for (int l = 0; l < 2; ++l) {
    const float* xin = (l == 0) ? x0 : xbuf;
    k_coeffs<<<gN, TB, 0, stream>>>(off_row, hp, ap, l, bdia, jacW, boffW);
    k_init<<<gNH, TB, 0, stream>>>(xin, Y, outbuf, c0p, l);
    for (int j = 0; j < RRC; ++j) {
      k_bj<<<gW, TB, 0, stream>>>((const float4*)Y, (float4*)BJ, boffW, bdia, off_row, adj_row);
      // 4 Jacobi sweeps: BJ->YK2->Y->YK2->Y  (KK even; result lands in Y)
      k_jacobi<<<gW, TB, 0, stream>>>((const float4*)BJ,  (float4*)YK2, (const float4*)BJ, jacW, off_col, adj_col);
      k_jacobi<<<gW, TB, 0, stream>>>((const float4*)YK2, (float4*)Y,   (const float4*)BJ, jacW, off_col, adj_col);
      k_jacobi<<<gW, TB, 0, stream>>>((const float4*)Y,   (float4*)YK2, (const float4*)BJ, jacW, off_col, adj_col);
      k_jacobi<<<gW, TB, 0, stream>>>((const float4*)YK2, (float4*)Y,   (const float4*)BJ, jacW, off_col, adj_col);
      k_upd<<<gNH, TB, 0, stream>>>(Y, outbuf, cjp, l, j);
    }
    k_relu<<<gNH, TB, 0, stream>>>(outbuf, xbuf);
  }

  // --- TopK pooling + mean + linear ---
  k_wnorm<<<1, 64, 0, stream>>>(topkw, invn);
  k_score<<<gW, TB, 0, stream>>>(xbuf, topkw, invn, sbuf, outbuf);
  k_thresh<<<GG, 512, 0, stream>>>(sbuf, thr);
  hipMemsetAsync(pooled, 0, (size_t)16 * HH * 4, stream);
  k_padw<<<4, TB, 0, stream>>>(linw, wpad);
  k_pool<<<GG, TB, 0, stream>>>(outbuf, sbuf, thr, pooled);
  k_lin<<<1, 32, 0, stream>>>(pooled, wpad, linb, out);
}